// VQ_17394617549038
// MI455X (gfx1250) — compile-verified
//
#include <hip/hip_runtime.h>
#include <stdint.h>

typedef float    v2f __attribute__((ext_vector_type(2)));
typedef float    v8f __attribute__((ext_vector_type(8)));
typedef uint32_t v4u __attribute__((ext_vector_type(4)));
typedef int      v8i __attribute__((ext_vector_type(8)));
typedef int      v4i __attribute__((ext_vector_type(4)));

#define D_DIM   256
#define K_CODES 1024
#define HW_SZ   1024
#define N_VEC   32768
#define MT      64        // z rows per workgroup

// ---------------------------------------------------------------------------
// ||e_k||^2 for each codebook row (K=1024, rows contiguous)
// ---------------------------------------------------------------------------
__global__ __launch_bounds__(256) void enorm_kernel(const float* __restrict__ cb,
                                                    float* __restrict__ enorm) {
    int k = blockIdx.x * 256 + threadIdx.x;
    const float4* row = (const float4*)(cb + (size_t)k * D_DIM);
    float s = 0.0f;
#pragma unroll 4
    for (int i = 0; i < D_DIM / 4; ++i) {
        float4 v = row[i];
        s += v.x * v.x + v.y * v.y + v.z * v.z + v.w * v.w;
    }
    enorm[k] = s;
}

// ---------------------------------------------------------------------------
// ||z_n||^2 per latent vector; stride HW over d, coalesced across threads.
// ---------------------------------------------------------------------------
__global__ __launch_bounds__(256) void znorm_kernel(const float* __restrict__ ze,
                                                    float* __restrict__ znorm) {
    int n = blockIdx.x * 256 + threadIdx.x;
    int b = n >> 10, hw = n & (HW_SZ - 1);
    const float* p = ze + (size_t)b * D_DIM * HW_SZ + hw;
    float s = 0.0f;
#pragma unroll 4
    for (int d = 0; d < D_DIM; ++d) {
        float v = p[(size_t)d * HW_SZ];
        s += v * v;
    }
    znorm[n] = s;
}

// ---------------------------------------------------------------------------
// Distance GEMM + argmin.  score(n,k) = ||e_k||^2 - 2 * z_n . e_k
// z tile (256 d x 64 rows, row stride HW) staged into LDS by the Tensor Data
// Mover (one 2D-D# tensor_load_to_lds from wave 0), WMMA f32 16x16x4 GEMM,
// per-lane running argmin, single cross-lane reduction at the end.
// ---------------------------------------------------------------------------
__global__ __launch_bounds__(256) void vq_argmin_kernel(const float* __restrict__ ze,
                                                        const float* __restrict__ cb,
                                                        const float* __restrict__ enorm,
                                                        float* __restrict__ rmin,
                                                        int* __restrict__ bidx) {
    __shared__ float zs[D_DIM][MT];   // 64 KB, written by TDM as [d][64]

    const int tid = threadIdx.x;
    const int n0  = blockIdx.x * MT;          // 64 | 1024 -> tile stays in one b
    const int b   = n0 >> 10;
    const int hw0 = n0 & (HW_SZ - 1);
    const float* zb = ze + (size_t)b * D_DIM * HW_SZ + hw0;

    // ---- TDM: tile_dim0=64 floats contiguous (X), tile_dim1=256 rows (Y),
    //      tensor_dim0_stride = 1024 floats; data_size=4B; dest = zs.
    if (tid < 32) {
        uint64_t gaddr  = (uint64_t)(uintptr_t)zb;
        uint32_t ldsoff = (uint32_t)(uintptr_t)&zs[0][0];
        v4u g0 = { 0x1u,                                   // count=1, user D#
                   ldsoff,                                 // lds_addr
                   (uint32_t)gaddr,                        // global_addr[31:0]
                   (uint32_t)(gaddr >> 32) | (2u << 30) }; // addr[56:32] | type=2
        v8i g1 = { 0x00020000,          // data_size=2 (4B), wg_mask=0
                   64  << 16,           // tensor_dim0 = 64 (lo16 in [63:48])
                   256 << 16,           // tensor_dim1 = 256 (lo16 in [95:80])
                   64  << 16,           // tile_dim0 = 64 ([127:112])
                   256,                 // tile_dim1 = 256 ([143:128]); tile_dim2=0
                   1024,                // tensor_dim0_stride = 1024 ([191:160])
                   0, 0 };
        v4i gz4 = { 0, 0, 0, 0 };
        v8i gz8 = { 0, 0, 0, 0, 0, 0, 0, 0 };
        __builtin_amdgcn_tensor_load_to_lds(g0, g1, gz4, gz4, gz8, 0);
        __builtin_amdgcn_s_wait_tensorcnt((short)0);
    }
    __syncthreads();

    const int wave     = tid >> 5;
    const int lane     = tid & 31;
    const int half     = lane >> 4;           // K-half of the x4 fragment
    const int l16      = lane & 15;           // M (A) / N (B) position
    const int rowbase  = (wave & 3) << 4;     // 0,16,32,48
    const int codebase = (wave >> 2) << 9;    // 0 or 512

    float bestd[8];
    int   besti[8];
#pragma unroll
    for (int v = 0; v < 8; ++v) { bestd[v] = 3.4e38f; besti[v] = 0; }

    for (int g = 0; g < 8; ++g) {             // 8 groups of 64 codes
        const int code0 = codebase + (g << 6);
        v8f acc[4] = { v8f{}, v8f{}, v8f{}, v8f{} };
        // B fragment: lane holds e_{code+l16}[k0+2*half : +1] -> contiguous pair
        const float* cbp = cb + (size_t)(code0 + l16) * D_DIM + (half << 1);

        for (int k0 = 0; k0 < D_DIM; k0 += 4) {
            v2f a;  // A 16x4 fp32 layout: VGPR0=K(2*half), VGPR1=K(2*half+1), M=l16
            a.x = zs[k0 + (half << 1)][rowbase + l16];
            a.y = zs[k0 + (half << 1) + 1][rowbase + l16];
            v2f b0 = *(const v2f*)(cbp + k0);
            v2f b1 = *(const v2f*)(cbp + 16 * D_DIM + k0);
            v2f b2 = *(const v2f*)(cbp + 32 * D_DIM + k0);
            v2f b3 = *(const v2f*)(cbp + 48 * D_DIM + k0);
            acc[0] = __builtin_amdgcn_wmma_f32_16x16x4_f32(false, a, false, b0, (short)0, acc[0], false, false);
            acc[1] = __builtin_amdgcn_wmma_f32_16x16x4_f32(false, a, false, b1, (short)0, acc[1], false, false);
            acc[2] = __builtin_amdgcn_wmma_f32_16x16x4_f32(false, a, false, b2, (short)0, acc[2], false, false);
            acc[3] = __builtin_amdgcn_wmma_f32_16x16x4_f32(false, a, false, b3, (short)0, acc[3], false, false);
        }

        // C/D layout: VGPR v, lanes 0-15 -> M=v, lanes 16-31 -> M=v+8; N=l16.
        // Per-lane running min only (branchless); codes ascend per lane, so
        // strict < keeps the lowest code index on ties.
#pragma unroll
        for (int j = 0; j < 4; ++j) {
            int   code = code0 + (j << 4) + l16;
            float en   = enorm[code];
#pragma unroll
            for (int v = 0; v < 8; ++v) {
                float dsc = __builtin_fmaf(-2.0f, acc[j][v], en);
                bool  t   = dsc < bestd[v];
                bestd[v] = t ? dsc  : bestd[v];
                besti[v] = t ? code : besti[v];
            }
        }
    }

    // one cross-lane min+index reduction per row (16 column classes)
#pragma unroll
    for (int v = 0; v < 8; ++v) {
        float dsc = bestd[v];
        int   di  = besti[v];
#pragma unroll
        for (int m = 1; m <= 8; m <<= 1) {
            float od = __shfl_xor(dsc, m, 32);
            int   oi = __shfl_xor(di,  m, 32);
            bool  t  = (od < dsc) || (od == dsc && oi < di);
            dsc = t ? od : dsc;
            di  = t ? oi : di;
        }
        bestd[v] = dsc;
        besti[v] = di;
    }

    // merge the two code-halves; reuse zs (all WMMA reads done after barrier)
    __syncthreads();
    float* mdist = &zs[0][0];         // [2][64]
    int*   midx  = (int*)&zs[4][0];   // [2][64]
    const int whalf = wave >> 2;
#pragma unroll
    for (int v = 0; v < 8; ++v) {
        if (l16 == v) {
            int r = rowbase + v + (half << 3);
            mdist[whalf * MT + r] = bestd[v];
            midx [whalf * MT + r] = besti[v];
        }
    }
    __syncthreads();
    if (tid < MT) {
        float d0 = mdist[tid], d1 = mdist[MT + tid];
        int   i0 = midx[tid],  i1 = midx[MT + tid];
        bool take1 = d1 < d0;                  // tie -> half0 (lower code index)
        rmin[n0 + tid] = take1 ? d1 : d0;
        bidx[n0 + tid] = take1 ? i1 : i0;
    }
}

// ---------------------------------------------------------------------------
// z_q[b,d,hw] = cb[idx[n]][d]  (hw-coalesced stores; codebook gathers hit L2)
// also emits the index output on the d==0 plane.
// ---------------------------------------------------------------------------
__global__ __launch_bounds__(256) void gather_kernel(const float* __restrict__ cb,
                                                     const int* __restrict__ bidx,
                                                     float* __restrict__ zq,
                                                     float* __restrict__ idxout) {
    int e  = blockIdx.x * 256 + threadIdx.x;
    int hw = e & (HW_SZ - 1);
    int d  = (e >> 10) & (D_DIM - 1);
    int b  = e >> 18;
    int n  = (b << 10) + hw;
    int idx = bidx[n];
    zq[e] = cb[(size_t)idx * D_DIM + d];
    if (d == 0) idxout[n] = (float)idx;
}

// ---------------------------------------------------------------------------
// loss = 1.25 * mean(||z_n||^2 + min_k score) over N*D elements
// ---------------------------------------------------------------------------
__global__ __launch_bounds__(256) void loss_kernel(const float* __restrict__ znorm,
                                                   const float* __restrict__ rmin,
                                                   float* __restrict__ lossout) {
    __shared__ float red[256];
    int tid = threadIdx.x;
    float s = 0.0f;
    for (int n = tid; n < N_VEC; n += 256) s += znorm[n] + rmin[n];
    red[tid] = s;
    __syncthreads();
    for (int off = 128; off > 0; off >>= 1) {
        if (tid < off) red[tid] += red[tid + off];
        __syncthreads();
    }
    if (tid == 0) lossout[0] = 1.25f * red[0] / 8388608.0f;
}

extern "C" void kernel_launch(void* const* d_in, const int* in_sizes, int n_in,
                              void* d_out, int out_size, void* d_ws, size_t ws_size,
                              hipStream_t stream) {
    (void)in_sizes; (void)n_in; (void)out_size; (void)ws_size;
    const float* ze = (const float*)d_in[0];   // [32,256,32,32]
    const float* cb = (const float*)d_in[1];   // [1024,256]

    float* out     = (float*)d_out;
    float* zq      = out;                      // 8388608 floats
    float* idxout  = out + (size_t)32 * 256 * 32 * 32;
    float* lossout = idxout + N_VEC;

    float* enorm = (float*)d_ws;               // 1024
    float* znorm = enorm + K_CODES;            // 32768
    float* rmin  = znorm + N_VEC;              // 32768
    int*   bidx  = (int*)(rmin + N_VEC);       // 32768

    enorm_kernel<<<K_CODES / 256, 256, 0, stream>>>(cb, enorm);
    znorm_kernel<<<N_VEC / 256, 256, 0, stream>>>(ze, znorm);
    vq_argmin_kernel<<<N_VEC / MT, 256, 0, stream>>>(ze, cb, enorm, rmin, bidx);
    gather_kernel<<<(32 * 256 * 32 * 32) / 256, 256, 0, stream>>>(cb, bidx, zq, idxout);
    loss_kernel<<<1, 256, 0, stream>>>(znorm, rmin, lossout);
}